// SpLearner_18837726560673
// MI455X (gfx1250) — compile-verified
//
#include <hip/hip_runtime.h>
#include <hip/hip_bf16.h>
#include <math.h>

typedef __attribute__((ext_vector_type(2))) float v2f;
typedef __attribute__((ext_vector_type(8))) float v8f;

#define DEG    16          // edges per node
#define FDIM   64          // feature dim
#define HDIM   128         // hidden dim (= 2*FDIM input to layer 1)
#define LDSROW 130         // padded LDS row stride (floats, even -> 8B aligned float2)

// ---------------------------------------------------------------------------
// Kernel 1: one wave (32 threads) per node = one 16-edge WMMA tile.
//   temp[16][128] = [feat[row] | feat[col_m]]
//   h1 = relu(temp @ W1^T + b1); h2 = relu(h1 @ W2^T + b2); z = h2 @ W3^T + b3
// Uses V_WMMA_F32_16X16X4_F32 (full f32 precision, K=4 per op, 32 K-steps).
// Writes z[16] to z_out (== d_out) and per-node sum(z^2) to blocksum.
// ---------------------------------------------------------------------------
__global__ __launch_bounds__(32) void splearner_mlp_kernel(
    const float* __restrict__ feat,
    const int*   __restrict__ cols,      // indices row 1 (cols), length E
    const float* __restrict__ W1, const float* __restrict__ b1,
    const float* __restrict__ W2, const float* __restrict__ b2,
    const float* __restrict__ W3, const float* __restrict__ b3,
    float* __restrict__ z_out,
    float* __restrict__ blocksum)
{
    __shared__ __align__(16) float hlds[DEG * LDSROW];  // 16 x 128 activations (padded)
    __shared__ float zbuf[DEG];

    const int row = blockIdx.x;
    const int l   = threadIdx.x;
    const int m   = l & 15;      // edge-in-tile (M index)
    const int hi  = l >> 4;      // lane half: K offset 2*hi (A & B frag layout)
    const int e0  = row * DEG;

    const int col = cols[e0 + m];

    // ---- A fragments for layer 1 (16x128 striped into 32 K-fragments) ----
    // ISA 32-bit A 16x4 layout: lane m holds row M=m; vgpr v, half hi -> k = v + 2*hi.
    v2f a[32];
    {
        const float* frow = feat + (size_t)row * FDIM;
        const float* fcol = feat + (size_t)col * FDIM;
#pragma unroll
        for (int ks = 0; ks < 16; ++ks)                   // k in [0,64): source-node feats
            a[ks] = *(const v2f*)(frow + ks * 4 + 2 * hi);
#pragma unroll
        for (int ks = 0; ks < 16; ++ks)                   // k in [64,128): neighbor feats
            a[16 + ks] = *(const v2f*)(fcol + ks * 4 + 2 * hi);
    }

    // ---- Layer 1: h = relu(temp @ W1^T + b1) ----
    for (int nt = 0; nt < HDIM / 16; ++nt) {
        const int n = nt * 16 + m;                        // output column handled by lane
        const float* wb = W1 + (size_t)n * HDIM + 2 * hi; // B[k][n] = W1[n][k]
        v8f c = {};
#pragma unroll
        for (int ks = 0; ks < 32; ++ks) {
            v2f b = *(const v2f*)(wb + ks * 4);
            c = __builtin_amdgcn_wmma_f32_16x16x4_f32(
                    false, a[ks], false, b, (short)0, c, false, false);
        }
        const float bias = b1[n];
        // C/D layout: lane -> column n, vgpr v -> row m = v + 8*hi
#pragma unroll
        for (int v = 0; v < 8; ++v) {
            float hv = c[v] + bias;
            hlds[(v + 8 * hi) * LDSROW + n] = fmaxf(hv, 0.0f);
        }
    }
    __syncthreads();

    // ---- Reload A fragments (h1) from LDS ----
#pragma unroll
    for (int ks = 0; ks < 32; ++ks)
        a[ks] = *(const v2f*)(&hlds[m * LDSROW + ks * 4 + 2 * hi]);
    __syncthreads();

    // ---- Layer 2: h2 = relu(h1 @ W2^T + b2) ----
    for (int nt = 0; nt < HDIM / 16; ++nt) {
        const int n = nt * 16 + m;
        const float* wb = W2 + (size_t)n * HDIM + 2 * hi;
        v8f c = {};
#pragma unroll
        for (int ks = 0; ks < 32; ++ks) {
            v2f b = *(const v2f*)(wb + ks * 4);
            c = __builtin_amdgcn_wmma_f32_16x16x4_f32(
                    false, a[ks], false, b, (short)0, c, false, false);
        }
        const float bias = b2[n];
#pragma unroll
        for (int v = 0; v < 8; ++v) {
            float hv = c[v] + bias;
            hlds[(v + 8 * hi) * LDSROW + n] = fmaxf(hv, 0.0f);
        }
    }
    __syncthreads();

    // ---- Layer 3: z[m] = h2[m] . W3 + b3 (16 lanes, 128 MACs each) ----
    if (l < DEG) {
        float acc = b3[0];
#pragma unroll 8
        for (int k = 0; k < HDIM; ++k)
            acc = fmaf(hlds[l * LDSROW + k], W3[k], acc);
        z_out[e0 + l] = acc;
        zbuf[l] = acc * acc;
    }
    __syncthreads();

    if (l == 0) {
        float s = 0.0f;
#pragma unroll
        for (int j = 0; j < DEG; ++j) s += zbuf[j];
        blocksum[row] = s;   // deterministic per-node partial (no float atomics)
    }
}

// ---------------------------------------------------------------------------
// Kernel 2: deterministic fixed-order reduction of per-node sum(z^2).
// ---------------------------------------------------------------------------
__global__ __launch_bounds__(256) void splearner_reduce_kernel(
    const float* __restrict__ blocksum, float* __restrict__ ss, int n)
{
    __shared__ float sh[256];
    const int t = threadIdx.x;
    float p = 0.0f;
    for (int i = t; i < n; i += 256) p += blocksum[i];   // fixed order per lane
    sh[t] = p;
    __syncthreads();
    for (int s = 128; s > 0; s >>= 1) {
        if (t < s) sh[t] += sh[t + s];
        __syncthreads();
    }
    if (t == 0) ss[0] = sh[0];
}

// ---------------------------------------------------------------------------
// Kernel 3: one thread per node. Normalize z, row-softmax over 16 edges,
// threshold at the 8th-largest pi (K=0.5 of deg 16), mask exactly like the
// reference: keep iff (pi - thre + 1e-15) > 0. In-place over d_out (each
// thread reads then rewrites only its own 16 contiguous slots).
// ---------------------------------------------------------------------------
__global__ __launch_bounds__(256) void splearner_topk_kernel(
    float* __restrict__ zio, const float* __restrict__ ss, int n)
{
    const int row = blockIdx.x * blockDim.x + threadIdx.x;
    if (row >= n) return;

    float nrm = sqrtf(ss[0]);
    nrm = fmaxf(nrm, 1e-12f);
    const float inv = 1.0f / nrm;

    const int e0 = row * DEG;
    float z[DEG];
#pragma unroll
    for (int j = 0; j < DEG; ++j) z[j] = zio[e0 + j] * inv;

    float zmax = z[0];
#pragma unroll
    for (int j = 1; j < DEG; ++j) zmax = fmaxf(zmax, z[j]);

    float ez[DEG], s = 0.0f;
#pragma unroll
    for (int j = 0; j < DEG; ++j) { ez[j] = expf(z[j] - zmax); s += ez[j]; }
    const float invs = 1.0f / s;

    float pi[DEG];
#pragma unroll
    for (int j = 0; j < DEG; ++j) pi[j] = ez[j] * invs;

    // 8th-largest (with multiplicity): 8 rounds of select-max (pi > 0 always)
    float t[DEG];
#pragma unroll
    for (int j = 0; j < DEG; ++j) t[j] = pi[j];
    float thre = 0.0f;
#pragma unroll
    for (int it = 0; it < DEG / 2; ++it) {
        float mx = -1.0f; int mi = 0;
#pragma unroll
        for (int j = 0; j < DEG; ++j)
            if (t[j] > mx) { mx = t[j]; mi = j; }
        thre = mx;
        t[mi] = -1.0f;
    }

#pragma unroll
    for (int j = 0; j < DEG; ++j) {
        const float p = pi[j];
        zio[e0 + j] = ((p - thre + 1e-15f) > 0.0f) ? p : 0.0f;
    }
}

// ---------------------------------------------------------------------------
extern "C" void kernel_launch(void* const* d_in, const int* in_sizes, int n_in,
                              void* d_out, int out_size, void* d_ws, size_t ws_size,
                              hipStream_t stream) {
    const float* feat    = (const float*)d_in[0];   // [N, 64]
    const int*   indices = (const int*)  d_in[1];   // [2, E] flat: rows then cols
    const float* W1 = (const float*)d_in[2];
    const float* b1 = (const float*)d_in[3];
    const float* W2 = (const float*)d_in[4];
    const float* b2 = (const float*)d_in[5];
    const float* W3 = (const float*)d_in[6];
    const float* b3 = (const float*)d_in[7];

    const int N = in_sizes[0] / FDIM;               // 50000
    const int E = N * DEG;                          // 800000
    const int* cols = indices + E;                  // second row of indices

    float* out      = (float*)d_out;                // z then masked pi, in place
    float* blocksum = (float*)d_ws;                 // [N]
    float* ss       = blocksum + N;                 // [1]

    splearner_mlp_kernel<<<N, 32, 0, stream>>>(
        feat, cols, W1, b1, W2, b2, W3, b3, out, blocksum);
    splearner_reduce_kernel<<<1, 256, 0, stream>>>(blocksum, ss, N);
    splearner_topk_kernel<<<(N + 255) / 256, 256, 0, stream>>>(out, ss, N);
}